// scGNN_24610162606563
// MI455X (gfx1250) — compile-verified
//
#include <hip/hip_runtime.h>
#include <math.h>

typedef __attribute__((ext_vector_type(2))) float v2f;
typedef __attribute__((ext_vector_type(8))) float v8f;

// ---------------- degree / normalization ----------------

__global__ void k_init_deg(float* __restrict__ deg, int N) {
    int i = blockIdx.x * blockDim.x + threadIdx.x;
    if (i < N) deg[i] = 1.0f;                    // self-loop contributes 1
}

__global__ void k_deg_scatter(const int* __restrict__ dst, float* __restrict__ deg, int E) {
    int e = blockIdx.x * blockDim.x + threadIdx.x;
    if (e < E) atomicAdd(&deg[dst[e]], 1.0f);
}

__global__ void k_rsqrt_inplace(float* __restrict__ deg, int N) {
    int i = blockIdx.x * blockDim.x + threadIdx.x;
    if (i < N) {
        float d = deg[i];
        deg[i] = (d > 0.0f) ? rsqrtf(d) : 0.0f;
    }
}

// ---------------- fp32 WMMA GEMM: C[M,NCOL] = A[M,K] @ W[K,NCOL] ----------------
// One wave32 computes one 16x16 tile of C using V_WMMA_F32_16X16X4_F32.
// A-frag (16x4 f32): lane L holds row (L&15), K = 2*(L>>4) + {vgpr0, vgpr1}.
// B-frag (4x16 f32): lane L holds col (L&15), K = 2*(L>>4) + {vgpr0, vgpr1}.
// C/D  (16x16 f32): vgpr v, lane L -> (M = v + 8*(L>>4), N = L&15).

template <int K, int NCOL>
__global__ void k_gemm_wmma(const float* __restrict__ A,
                            const float* __restrict__ W,
                            float* __restrict__ C, int M) {
    const int lane  = threadIdx.x & 31;
    const int wib   = threadIdx.x >> 5;
    const int wpb   = blockDim.x >> 5;
    const int gwave = blockIdx.x * wpb + wib;

    constexpr int NT = NCOL / 16;
    const int tm = gwave / NT;
    const int tn = gwave % NT;
    if (tm * 16 >= M) return;                    // wave-uniform guard (EXEC stays all-1s)

    const int hi  = lane >> 4;                   // 0 or 1
    const int lm  = lane & 15;
    int row = tm * 16 + lm;
    if (row >= M) row = M - 1;                   // clamp (keeps EXEC full for WMMA)
    const int col = tn * 16 + lm;

    const float* __restrict__ arow = A + (size_t)row * K;

    v8f acc = {};
#pragma unroll
    for (int k0 = 0; k0 < K; k0 += 4) {
        const int kk = k0 + 2 * hi;
        v2f a, b;
        a.x = arow[kk];
        a.y = arow[kk + 1];
        b.x = W[(size_t)kk * NCOL + col];
        b.y = W[(size_t)(kk + 1) * NCOL + col];
        acc = __builtin_amdgcn_wmma_f32_16x16x4_f32(
            /*neg_a=*/false, a, /*neg_b=*/false, b,
            /*c_mod=*/(short)0, acc, /*reuse_a=*/false, /*reuse_b=*/false);
    }

    float* __restrict__ crow = C + (size_t)(tm * 16) * NCOL + (size_t)tn * 16;
#pragma unroll
    for (int v = 0; v < 8; ++v) {
        const int m = v + 8 * hi;
        if (tm * 16 + m < M)
            crow[(size_t)m * NCOL + lm] = acc[v];
    }
}

// ---------------- aggregation ----------------

// out[i*F+f] = h[i*F+f] * dinv[i]^2 + bias[f]   (self-loop term + bias)
template <int F>
__global__ void k_init_agg(const float* __restrict__ h,
                           const float* __restrict__ dinv,
                           const float* __restrict__ bias,
                           float* __restrict__ out, long long total) {
    long long t = (long long)blockIdx.x * blockDim.x + threadIdx.x;
    if (t >= total) return;
    const int i = (int)(t / F);
    const int f = (int)(t % F);
    const float di = dinv[i];
    out[t] = h[t] * di * di + bias[f];
}

// one thread per (edge, feature): out[dst*F+f] += h[src*F+f] * dinv[src]*dinv[dst]
template <int F>
__global__ void k_edge_scatter(const int* __restrict__ src,
                               const int* __restrict__ dst,
                               const float* __restrict__ dinv,
                               const float* __restrict__ h,
                               float* __restrict__ out, int E) {
    long long t = (long long)blockIdx.x * blockDim.x + threadIdx.x;
    const int e = (int)(t / F);
    const int f = (int)(t % F);
    if (e >= E) return;
    const int s = src[e];
    const int d = dst[e];
    const float nrm = dinv[s] * dinv[d];
    atomicAdd(&out[(long long)d * F + f], h[(long long)s * F + f] * nrm);
}

__global__ void k_relu_inplace(float* __restrict__ x, long long total) {
    long long t = (long long)blockIdx.x * blockDim.x + threadIdx.x;
    if (t < total) x[t] = fmaxf(x[t], 0.0f);
}

// ---------------- launcher ----------------

extern "C" void kernel_launch(void* const* d_in, const int* in_sizes, int n_in,
                              void* d_out, int out_size, void* d_ws, size_t ws_size,
                              hipStream_t stream) {
    const float* x  = (const float*)d_in[0];   // [N, 64]
    const int*   ei = (const int*)  d_in[1];   // [2, E]
    const float* W1 = (const float*)d_in[2];   // [64, 32]
    const float* b1 = (const float*)d_in[3];   // [32]
    const float* W2 = (const float*)d_in[4];   // [32, 64]
    const float* b2 = (const float*)d_in[5];   // [64]
    float* out = (float*)d_out;                // [N, 64]

    const int N = in_sizes[0] / 64;
    const int E = in_sizes[1] / 2;
    const int* src = ei;
    const int* dst = ei + E;

    // workspace layout (floats): dinv(N) | h1(32N) | agg1(32N) | h2(64N)
    float* ws   = (float*)d_ws;
    float* dinv = ws;
    float* h1   = dinv + N;
    float* agg1 = h1 + (size_t)N * 32;
    float* h2   = agg1 + (size_t)N * 32;

    const int T = 256;
    const int gN = (N + T - 1) / T;
    const int gE = (E + T - 1) / T;

    // 1) symmetric normalization coefficients
    k_init_deg<<<gN, T, 0, stream>>>(dinv, N);
    k_deg_scatter<<<gE, T, 0, stream>>>(dst, dinv, E);
    k_rsqrt_inplace<<<gN, T, 0, stream>>>(dinv, N);

    // 2) layer 1: H1 = X @ W1   (N x 64 @ 64 x 32) via fp32 WMMA
    {
        const int tiles  = ((N + 15) / 16) * (32 / 16);
        const int blocks = (tiles + 3) / 4;            // 4 waves per 128-thread block
        k_gemm_wmma<64, 32><<<blocks, 128, 0, stream>>>(x, W1, h1, N);
    }
    const long long tot1 = (long long)N * 32;
    k_init_agg<32><<<(int)((tot1 + T - 1) / T), T, 0, stream>>>(h1, dinv, b1, agg1, tot1);
    {
        const long long et = (long long)E * 32;
        k_edge_scatter<32><<<(int)((et + T - 1) / T), T, 0, stream>>>(src, dst, dinv, h1, agg1, E);
    }
    k_relu_inplace<<<(int)((tot1 + T - 1) / T), T, 0, stream>>>(agg1, tot1);

    // 3) layer 2: H2 = relu(agg1) @ W2   (N x 32 @ 32 x 64) via fp32 WMMA
    {
        const int tiles  = ((N + 15) / 16) * (64 / 16);
        const int blocks = (tiles + 3) / 4;
        k_gemm_wmma<32, 64><<<blocks, 128, 0, stream>>>(agg1, W2, h2, N);
    }
    const long long tot2 = (long long)N * 64;
    k_init_agg<64><<<(int)((tot2 + T - 1) / T), T, 0, stream>>>(h2, dinv, b2, out, tot2);
    {
        const long long et = (long long)E * 64;
        k_edge_scatter<64><<<(int)((et + T - 1) / T), T, 0, stream>>>(src, dst, dinv, h2, out, E);
    }
}